// FeatTimeAttention_29832842838365
// MI455X (gfx1250) — compile-verified
//
#include <hip/hip_runtime.h>
#include <math.h>

// Problem constants (from reference)
#define BS    64
#define T     128
#define DF    64
#define UNITS 128

// LDS pitches chosen for bank-conflict avoidance (64 banks x 4B)
#define FP_PITCH 132   // (4*row + col) % 64 -> 16 distinct banks for column fetches
#define G_PITCH  66    // 64 gram cols + col 64 = x_ty + 1 pad

typedef __attribute__((ext_vector_type(2))) float v2f;
typedef __attribute__((ext_vector_type(8))) float v8f;

// One workgroup (256 threads = 8 waves, wave32) per (b,t) pair.
__global__ __launch_bounds__(256) void fta_per_bt_kernel(
    const float* __restrict__ x,       // (BS,T,DF)
    const float* __restrict__ latent,  // (BS,T,UNITS)
    const float* __restrict__ kern,    // (DF,UNITS)
    const float* __restrict__ bias,    // (DF,UNITS)
    const float* __restrict__ beta,    // (T)
    float* __restrict__ ws,            // (BS,T,UNITS) scaled o_hat (or null)
    float* __restrict__ out,           // (BS,UNITS) when use_atomic
    int use_atomic)
{
  __shared__ float fp[DF * FP_PITCH];   // 64 x 128 feature projections
  __shared__ float G[DF * G_PITCH];     // cols 0..63 = gram, col 64 = x_ty
  __shared__ float alpha[DF];
  __shared__ float bsh[T];              // |beta|+eps staging

  const int tid = threadIdx.x;
  const int t = blockIdx.x;
  const int b = blockIdx.y;

  if (tid < T) bsh[tid] = fabsf(beta[tid]) + 1e-8f;

  // ---- Step 1: fp[d][u] = x[b,t,d]*K[d][u] + B[d][u] ----
  const float* xbt = x + (size_t)(b * T + t) * DF;
  for (int idx = tid; idx < DF * UNITS; idx += 256) {
    const int d = idx >> 7;      // /UNITS
    const int u = idx & (UNITS - 1);
    fp[d * FP_PITCH + u] = xbt[d] * kern[idx] + bias[idx];
  }
  __syncthreads();

  // ---- Step 2: gram = fp @ fp^T via V_WMMA_F32_16X16X4_F32 ----
  // 4x4 grid of 16x16 tiles; 8 waves, 2 tiles each. EXEC is full here.
  {
    const int wave = tid >> 5;
    const int lane = tid & 31;
    const int g  = lane >> 4;    // half-wave group: K pair select
    const int mn = lane & 15;    // M (for A) / N (for B) index
    for (int tt = 0; tt < 2; ++tt) {
      const int tile = wave + tt * 8;           // 0..15
      const int ti = tile >> 2, tj = tile & 3;  // tile row/col
      const float* rowA = &fp[(16 * ti + mn) * FP_PITCH];
      const float* rowB = &fp[(16 * tj + mn) * FP_PITCH];
      v8f c = {};
      for (int k0 = 0; k0 < UNITS; k0 += 4) {
        const int ka = k0 + 2 * g;
        v2f a, bb;
        // A (16x4): lane holds fp[16ti+mn][ka], fp[16ti+mn][ka+1]
        a.x  = rowA[ka];  a.y  = rowA[ka + 1];
        // B (4x16) = fp^T: lane holds fp[16tj+mn][ka], fp[16tj+mn][ka+1]
        bb.x = rowB[ka];  bb.y = rowB[ka + 1];
        c = __builtin_amdgcn_wmma_f32_16x16x4_f32(
            /*neg_a=*/false, a, /*neg_b=*/false, bb,
            /*c_mod=*/(short)0, c, /*reuse_a=*/false, /*reuse_b=*/false);
      }
      // C/D layout: VGPR v, lane -> row = v + 8*g, col = mn
      for (int v = 0; v < 8; ++v)
        G[(16 * ti + v + 8 * g) * G_PITCH + (16 * tj + mn)] = c[v];
    }
  }

  // ---- Step 3: x_ty[d] = fp[d,:] . latent[b,t,:]  -> column 64 of G ----
  const float* ybt = latent + (size_t)(b * T + t) * UNITS;
  if (tid < DF) {
    float s = 0.f;
    for (int u = 0; u < UNITS; ++u) s += fp[tid * FP_PITCH + u] * ybt[u];
    G[tid * G_PITCH + DF] = s;
  }
  __syncthreads();

  // ---- Step 4: Gauss-Jordan on [G | x_ty] (SPD -> no pivoting) ----
  {
    const int r  = tid & 63;   // row handled by this thread
    const int cl = tid >> 6;   // 4 threads per row cover columns strided by 4
    for (int p = 0; p < DF; ++p) {
      const float inv = 1.0f / G[p * G_PITCH + p];
      if (r != p) {
        const float f = G[r * G_PITCH + p] * inv;
        for (int c = p + 1 + cl; c <= DF; c += 4)
          G[r * G_PITCH + c] -= f * G[p * G_PITCH + c];
      }
      __syncthreads();
    }
  }
  if (tid < DF)
    alpha[tid] = G[tid * G_PITCH + DF] / G[tid * G_PITCH + tid];
  __syncthreads();

  // ---- Step 5: o_hat[u] = alpha . fp[:,u], scaled by beta score ----
  if (tid < UNITS) {
    float S = 0.f;
    for (int k = 0; k < T; ++k) S += bsh[k];
    const float bscale = bsh[t] / S;
    float oh = 0.f;
    for (int d = 0; d < DF; ++d) oh += alpha[d] * fp[d * FP_PITCH + tid];
    const float val = oh * bscale;
    if (use_atomic) {
      atomicAdd(&out[b * UNITS + tid], val);
    } else {
      ws[(size_t)(b * T + t) * UNITS + tid] = val;
    }
  }
}

// Deterministic reduction over t: z[b,u] = sum_t ws[b,t,u]
__global__ __launch_bounds__(256) void fta_reduce_t_kernel(
    const float* __restrict__ ws, float* __restrict__ out)
{
  const int idx = blockIdx.x * 256 + threadIdx.x;  // 0..BS*UNITS-1
  if (idx >= BS * UNITS) return;
  const int b = idx / UNITS;
  const int u = idx - b * UNITS;
  const float* p = ws + (size_t)b * T * UNITS + u;
  float s = 0.f;
  for (int tt = 0; tt < T; ++tt) s += p[tt * UNITS];
  out[idx] = s;
}

__global__ __launch_bounds__(256) void fta_zero_kernel(float* __restrict__ p, int n)
{
  const int idx = blockIdx.x * 256 + threadIdx.x;
  if (idx < n) p[idx] = 0.f;
}

extern "C" void kernel_launch(void* const* d_in, const int* in_sizes, int n_in,
                              void* d_out, int out_size, void* d_ws, size_t ws_size,
                              hipStream_t stream) {
  const float* x      = (const float*)d_in[0];
  const float* latent = (const float*)d_in[1];
  const float* kern   = (const float*)d_in[2];
  const float* bias   = (const float*)d_in[3];
  const float* beta   = (const float*)d_in[4];
  float* out = (float*)d_out;

  const size_t need = (size_t)BS * T * UNITS * sizeof(float);
  dim3 grid(T, BS);  // one workgroup per (b,t)

  if (ws_size >= need) {
    float* ws = (float*)d_ws;
    fta_per_bt_kernel<<<grid, 256, 0, stream>>>(x, latent, kern, bias, beta,
                                                ws, out, 0);
    fta_reduce_t_kernel<<<(BS * UNITS + 255) / 256, 256, 0, stream>>>(ws, out);
  } else {
    // Fallback: accumulate with f32 atomics (zero output first).
    fta_zero_kernel<<<(BS * UNITS + 255) / 256, 256, 0, stream>>>(out, BS * UNITS);
    fta_per_bt_kernel<<<grid, 256, 0, stream>>>(x, latent, kern, bias, beta,
                                                nullptr, out, 1);
  }
}